// S4Layer_22204980920820
// MI455X (gfx1250) — compile-verified
//
#include <hip/hip_runtime.h>
#include <math.h>
#include <stdint.h>

// Shapes fixed by the harness setup_inputs():
#define BATCH  16
#define LSEQ   2048
#define DMODEL 512
#define NSTATE 64
#define ROWS   (BATCH * LSEQ)   // 32768
#define LN_EPS 1e-5f

typedef __attribute__((ext_vector_type(2))) float v2f;
typedef __attribute__((ext_vector_type(8))) float v8f;
typedef __attribute__((ext_vector_type(4))) int   v4i;

// ---------------------------------------------------------------------------
// Native tanh: builtin if available, else raw v_tanh_f32 (CDNA5 trans op).
// The 2048-long dependent tanh chain is the latency floor of the whole layer,
// so we must not fall back to the ~30-instruction ocml tanh expansion.
// ---------------------------------------------------------------------------
__device__ __forceinline__ float fast_tanh(float x) {
#if __has_builtin(__builtin_amdgcn_tanhf)
  return __builtin_amdgcn_tanhf(x);
#elif __has_builtin(__builtin_amdgcn_tanh_f32)
  return __builtin_amdgcn_tanh_f32(x);
#else
  float r;
  // v_nop respects the trans-op result hazard (1 op before use).
  asm volatile("v_tanh_f32 %0, %1\n\tv_nop" : "=v"(r) : "v"(x));
  return r;
#endif
}

// ---------------------------------------------------------------------------
// Async global->LDS 16-byte copy (GLOBAL_LOAD_ASYNC_TO_LDS_B128, ASYNCcnt).
// Builtin signature (decoded from hipcc diagnostics):
//   (AS1 v4i* gsrc, AS3 v4i* lds_dst, imm offset, imm cpol)
// ---------------------------------------------------------------------------
#if __has_builtin(__builtin_amdgcn_global_load_async_to_lds_b128)
#define HAS_ASYNC_LDS 1
typedef __attribute__((address_space(1))) v4i g_v4i_t;
typedef __attribute__((address_space(3))) v4i l_v4i_t;
__device__ __forceinline__ void async_copy16(void* lds_dst, const void* gsrc) {
  __builtin_amdgcn_global_load_async_to_lds_b128(
      (g_v4i_t*)gsrc, (l_v4i_t*)lds_dst, 0, 0);
}
__device__ __forceinline__ void async_wait0() {
#if __has_builtin(__builtin_amdgcn_s_wait_asynccnt)
  __builtin_amdgcn_s_wait_asynccnt(0);
#else
  asm volatile("s_wait_asynccnt 0x0" ::: "memory");
#endif
}
#else
#define HAS_ASYNC_LDS 0
#endif

// -----------------------------------------------------------------------------
// Kernel 0: one-shot transpose of weight A[512][64] -> At[64][512] so the
// WMMA B-fragment (two consecutive K values per lane) is one contiguous b64.
// -----------------------------------------------------------------------------
__global__ __launch_bounds__(256) void s4_transpose_A(
    const float* __restrict__ A, float* __restrict__ At) {
  int i = blockIdx.x * 256 + threadIdx.x;  // i < 512*64, coalesced read
  int k = i >> 6, n = i & 63;
  At[(size_t)n * DMODEL + k] = A[i];
}

// -----------------------------------------------------------------------------
// Kernel 1: fused LayerNorm + G = x_norm @ A
// Block: 128 thr = 4 waves, 32 rows x all N=64. Wave w owns N-tile [16w,16w+16)
// and runs TWO independent accumulator chains (rows 0-15 / 16-31) sharing each
// B-fragment -> halves weight traffic, back-to-back WMMA issue.
// -----------------------------------------------------------------------------
__global__ __launch_bounds__(128) void s4_ln_gemm1(
    const float* __restrict__ x, const float* __restrict__ At,
    const float* __restrict__ gamma, const float* __restrict__ beta,
    float* __restrict__ G, float* __restrict__ stats) {
  __shared__ float xs[32 * 516];  // 32 rows, stride 516 (pad -> bank spread)
  __shared__ float red_s[128], red_q[128];
  __shared__ float s_mean[32], s_rstd[32];

  const int tid = threadIdx.x;
  const long m0 = (long)blockIdx.x * 32;

  // --- stage 32 rows of x into LDS (async B128 when available) ---
  const float* xrow = x + (size_t)m0 * DMODEL;
#if HAS_ASYNC_LDS
  // 16384 floats = 4096 x 16B chunks; lane-consecutive chunks -> coalesced.
  for (int c = tid; c < 32 * DMODEL / 4; c += 128) {
    int r = c >> 7, col = (c & 127) * 4;
    async_copy16(&xs[r * 516 + col], &xrow[c * 4]);
  }
  async_wait0();
#else
  for (int i = tid; i < 32 * DMODEL; i += 128) {
    int r = i >> 9, c = i & (DMODEL - 1);
    xs[r * 516 + c] = xrow[i];
  }
#endif
  __syncthreads();

  // --- LayerNorm stats: 4 threads per row, 128 elements each ---
  {
    int r = tid >> 2, l4 = tid & 3;
    const float* rp = &xs[r * 516 + l4 * 128];
    float s = 0.f, q = 0.f;
    #pragma unroll 8
    for (int j = 0; j < 128; ++j) { float v = rp[j]; s += v; q += v * v; }
    red_s[tid] = s; red_q[tid] = q;
  }
  __syncthreads();
  if ((tid & 3) == 0) {
    int r = tid >> 2;
    float s = 0.f, q = 0.f;
    #pragma unroll
    for (int j = 0; j < 4; ++j) { s += red_s[tid + j]; q += red_q[tid + j]; }
    float mean = s * (1.0f / DMODEL);
    float var  = q * (1.0f / DMODEL) - mean * mean;
    float rstd = rsqrtf(var + LN_EPS);
    s_mean[r] = mean; s_rstd[r] = rstd;
    stats[(m0 + r) * 2 + 0] = mean;   // reused by the epilogue kernel
    stats[(m0 + r) * 2 + 1] = rstd;
  }
  __syncthreads();

  // --- normalize in LDS ---
  for (int i = tid; i < 32 * DMODEL; i += 128) {
    int r = i >> 9, c = i & (DMODEL - 1);
    float v = xs[r * 516 + c];
    xs[r * 516 + c] = (v - s_mean[r]) * s_rstd[r] * gamma[c] + beta[c];
  }
  __syncthreads();

  // --- WMMA f32 16x16x4: 128 K-steps, 2 WMMA per step sharing one B-frag ---
  const int lane = tid & 31;
  const int w    = tid >> 5;
  const int lm   = lane & 15;   // M for A-frag, N for B-frag / C-frag
  const int kh   = lane >> 4;   // K-half selector
  const int col  = w * 16 + lm; // output column in [0,64)

  const float* bt = At + (size_t)col * DMODEL;  // lane streams contiguous K
  v8f acc0 = {}, acc1 = {};
  #pragma unroll 4
  for (int k = 0; k < DMODEL; k += 4) {
    const int kk = k + 2 * kh;
    v2f b  = *(const v2f*)&bt[kk];
    v2f a0 = *(const v2f*)&xs[lm * 516 + kk];
    v2f a1 = *(const v2f*)&xs[(16 + lm) * 516 + kk];
    acc0 = __builtin_amdgcn_wmma_f32_16x16x4_f32(
        false, a0, false, b, (short)0, acc0, false, false);
    acc1 = __builtin_amdgcn_wmma_f32_16x16x4_f32(
        false, a1, false, b, (short)0, acc1, false, false);
  }

  // --- store G tiles (C/D layout: row = v + 8*kh, col = lm) ---
  float* Grow = G + (size_t)m0 * NSTATE;
  #pragma unroll
  for (int v = 0; v < 8; ++v) {
    int rr = v + 8 * kh;
    Grow[(size_t)rr * NSTATE + col]        = acc0[v];
    Grow[(size_t)(rr + 16) * NSTATE + col] = acc1[v];
  }
}

// -----------------------------------------------------------------------------
// Kernel 2: sequential scan h_t = tanh(G_t + h_{t-1}); 1024 independent (b,n)
// chains; 8-deep prefetch hides L2 latency behind the native-tanh dep chain.
// -----------------------------------------------------------------------------
__global__ __launch_bounds__(64) void s4_scan(
    const float* __restrict__ G, float* __restrict__ H) {
  const int b = blockIdx.x;
  const int n = threadIdx.x;
  const float* g = G + (size_t)b * LSEQ * NSTATE + n;
  float*       h = H + (size_t)b * LSEQ * NSTATE + n;

  float hv = 0.f;
  float nxt[8];
  #pragma unroll
  for (int j = 0; j < 8; ++j) nxt[j] = g[j * NSTATE];

  for (int t0 = 0; t0 < LSEQ; t0 += 8) {
    float cur[8];
    #pragma unroll
    for (int j = 0; j < 8; ++j) cur[j] = nxt[j];
    if (t0 + 8 < LSEQ) {
      #pragma unroll
      for (int j = 0; j < 8; ++j) nxt[j] = g[(t0 + 8 + j) * NSTATE];
    }
    #pragma unroll
    for (int j = 0; j < 8; ++j) {
      hv = fast_tanh(cur[j] + hv);
      h[(t0 + j) * NSTATE] = hv;
    }
  }
}

// -----------------------------------------------------------------------------
// Kernel 3: out = H @ C^T + x_norm * D + x
// Block: 128 thr = 4 waves, 32 rows x all 512 cols; two accumulator chains per
// wave share each C-fragment (halves C traffic). x_norm recomputed from stats.
// -----------------------------------------------------------------------------
__global__ __launch_bounds__(128) void s4_gemm2_epilogue(
    const float* __restrict__ H, const float* __restrict__ C,
    const float* __restrict__ x, const float* __restrict__ D,
    const float* __restrict__ gamma, const float* __restrict__ beta,
    const float* __restrict__ stats, float* __restrict__ out) {
  __shared__ float hs[32 * 68];   // 32 rows of H, padded stride 68
  __shared__ float s_mean[32], s_rstd[32];

  const int tid = threadIdx.x;
  const long m0 = (long)blockIdx.x * 32;

  // --- stage 32 rows of H into LDS ---
  const float* Hrow = H + (size_t)m0 * NSTATE;
#if HAS_ASYNC_LDS
  for (int c = tid; c < 32 * NSTATE / 4; c += 128) {   // 512 x 16B chunks
    int r = c >> 4, col = (c & 15) * 4;
    async_copy16(&hs[r * 68 + col], &Hrow[c * 4]);
  }
  async_wait0();
#else
  for (int i = tid; i < 32 * NSTATE; i += 128) {
    int r = i >> 6, c = i & 63;
    hs[r * 68 + c] = Hrow[i];
  }
#endif
  if (tid < 32) {
    s_mean[tid] = stats[(m0 + tid) * 2 + 0];
    s_rstd[tid] = stats[(m0 + tid) * 2 + 1];
  }
  __syncthreads();

  const int lane = tid & 31;
  const int w    = tid >> 5;
  const int lm   = lane & 15;
  const int kh   = lane >> 4;

  for (int nt = 0; nt < 8; ++nt) {
    const int d = (nt * 4 + w) * 16 + lm;   // output column in [0,512)
    v8f acc0 = {}, acc1 = {};
    #pragma unroll
    for (int k = 0; k < NSTATE; k += 4) {   // 16 K-steps, 2 WMMA each
      const int kk = k + 2 * kh;
      v2f b  = *(const v2f*)&C[(size_t)d * NSTATE + kk];  // C^T contiguous pair
      v2f a0 = *(const v2f*)&hs[lm * 68 + kk];
      v2f a1 = *(const v2f*)&hs[(16 + lm) * 68 + kk];
      acc0 = __builtin_amdgcn_wmma_f32_16x16x4_f32(
          false, a0, false, b, (short)0, acc0, false, false);
      acc1 = __builtin_amdgcn_wmma_f32_16x16x4_f32(
          false, a1, false, b, (short)0, acc1, false, false);
    }
    const float Dd = D[d], gd = gamma[d], bd = beta[d];
    #pragma unroll
    for (int v = 0; v < 8; ++v) {
      int rr0 = v + 8 * kh;
      int rr1 = rr0 + 16;
      size_t i0 = (size_t)(m0 + rr0) * DMODEL + d;
      size_t i1 = (size_t)(m0 + rr1) * DMODEL + d;
      float x0 = x[i0], x1 = x[i1];
      float xn0 = (x0 - s_mean[rr0]) * s_rstd[rr0] * gd + bd;
      float xn1 = (x1 - s_mean[rr1]) * s_rstd[rr1] * gd + bd;
      out[i0] = acc0[v] + xn0 * Dd + x0;
      out[i1] = acc1[v] + xn1 * Dd + x1;
    }
  }
}

// -----------------------------------------------------------------------------
extern "C" void kernel_launch(void* const* d_in, const int* in_sizes, int n_in,
                              void* d_out, int out_size, void* d_ws, size_t ws_size,
                              hipStream_t stream) {
  const float* x     = (const float*)d_in[0];
  const float* A     = (const float*)d_in[1];
  const float* C     = (const float*)d_in[2];
  const float* D     = (const float*)d_in[3];
  const float* gamma = (const float*)d_in[4];
  const float* beta  = (const float*)d_in[5];
  float* out = (float*)d_out;

  // ws layout: stats[ROWS*2] | At[64*512] | G[ROWS*64] | H[ROWS*64] (~16.6 MB)
  float* stats = (float*)d_ws;
  float* At    = stats + (size_t)ROWS * 2;
  float* G     = At    + (size_t)NSTATE * DMODEL;
  float* H     = G     + (size_t)ROWS * NSTATE;

  s4_transpose_A<<<(DMODEL * NSTATE) / 256, 256, 0, stream>>>(A, At);
  s4_ln_gemm1<<<ROWS / 32, 128, 0, stream>>>(x, At, gamma, beta, G, stats);
  s4_scan<<<BATCH, NSTATE, 0, stream>>>(G, H);
  s4_gemm2_epilogue<<<ROWS / 32, 128, 0, stream>>>(H, C, x, D, gamma, beta,
                                                   stats, out);
}